// UmbrellaSurfaceConstructor_14259291422781
// MI455X (gfx1250) — compile-verified
//
#include <hip/hip_runtime.h>
#include <math.h>

#define NPTS   65536
#define KNN    54
#define CHUNKC 1024
#define PI_F     3.14159265358979323846f
#define INV2PI_F 0.15915494309189535f

typedef __attribute__((ext_vector_type(2))) float        v2f;
typedef __attribute__((ext_vector_type(8))) float        v8f;
typedef __attribute__((ext_vector_type(4))) unsigned int v4u;
typedef __attribute__((ext_vector_type(4))) int          v4i;
typedef __attribute__((ext_vector_type(8))) int          v8i;

// ---------------------------------------------------------------------------
// Kernel 1: build augmented candidate vectors  B[c] = (-2x, -2y, -2z, |c|^2)
// so that  [qx,qy,qz,1] . B[c] = |q-c|^2 - |q|^2  (ranking-exact distance)
// ---------------------------------------------------------------------------
__global__ __launch_bounds__(256) void bvec_kernel(const float* __restrict__ xyz,
                                                   float4* __restrict__ bvec) {
    int i = blockIdx.x * blockDim.x + threadIdx.x;
    if (i >= NPTS) return;
    float x = xyz[3 * i], y = xyz[3 * i + 1], z = xyz[3 * i + 2];
    bvec[i] = make_float4(-2.f * x, -2.f * y, -2.f * z, x * x + y * y + z * z);
}

// ---------------------------------------------------------------------------
// TDM: DMA one 16KB candidate chunk (4096 dwords, 1-row tile) into LDS.
// D# built per CDNA5 ISA 8.3/8.4: group0 = {count=1 | lds_addr | global_addr |
// type=2}; group1 = {data_size=4B, tensor_dim0=4096, tensor_dim1=1,
// tile_dim0=4096, tile_dim1=1, tensor_dim0_stride=4096}. Groups 2/3 zero (2D).
// Tracked with TENSORcnt; issued by wave 0 only (EXEC is ignored by TDM).
// ---------------------------------------------------------------------------
__device__ __forceinline__ void tdm_load_chunk(const float4* gsrc, unsigned lds_addr) {
    unsigned long long ga = (unsigned long long)(uintptr_t)gsrc;
    v4u g0;
    g0[0] = 1u;                                        // count=1, user D#
    g0[1] = lds_addr;                                  // LDS byte address
    g0[2] = (unsigned)(ga & 0xFFFFFFFFull);            // global_addr[31:0]
    g0[3] = (unsigned)((ga >> 32) & 0x01FFFFFFull)     // global_addr[56:32]
          | (2u << 30);                                // type = 2 ("image")
    v8i g1;
    g1[0] = (int)(2u << 16);                           // data_size=4B (code 2)
    g1[1] = (int)(4096u << 16);                        // tensor_dim0[15:0]=4096
    g1[2] = (int)(1u << 16);                           // tensor_dim1[15:0]=1
    g1[3] = (int)(4096u << 16);                        // tile_dim0=4096
    g1[4] = 1;                                         // tile_dim1=1, tile_dim2=0
    g1[5] = 4096;                                      // tensor_dim0_stride
    g1[6] = 0;
    g1[7] = 0;
    v4i gz = {0, 0, 0, 0};
#if __clang_major__ >= 23
    v8i gz8 = {0, 0, 0, 0, 0, 0, 0, 0};
    __builtin_amdgcn_tensor_load_to_lds(g0, g1, gz, gz, gz8, 0);
#else
    __builtin_amdgcn_tensor_load_to_lds(g0, g1, gz, gz, 0);
#endif
}

// ---------------------------------------------------------------------------
// Kernel 2: brute-force KNN via V_WMMA_F32_16X16X4_F32.
// Block = 256 threads = 8 waves; each wave owns 32 queries (two 16-row A tiles).
// Candidate chunks are double-buffered in LDS via the Tensor Data Mover, so
// the DMA of chunk i+1 overlaps the WMMA + top-54 selection on chunk i.
// Per 16-candidate tile: 2 WMMAs -> 32x16 distance block -> LDS transpose ->
// each lane runs bounded-insertion top-54 for its own query.
// Output: 54 neighbor indices per query, sorted ascending by distance.
// ---------------------------------------------------------------------------
__global__ __launch_bounds__(256) void knn_kernel(const float* __restrict__ xyz,
                                                  const float4* __restrict__ bvec,
                                                  int* __restrict__ knn_out) {
    __shared__ float4 sB[2][CHUNKC];     // 2 x 16 KB candidate chunks (TDM dest)
    __shared__ float  sD[8][16][32];     // 16 KB: [wave][candidate n][query m]

    const int tid  = threadIdx.x;
    const int w    = tid >> 5;
    const int lane = tid & 31;
    const int hh   = lane >> 4;          // which half of the wave
    const int lr   = lane & 15;
    const int qbase = blockIdx.x * 256 + w * 32;

    const unsigned lds_b0 = (unsigned)(unsigned long long)(uintptr_t)&sB[0][0];
    const unsigned lds_b1 = (unsigned)(unsigned long long)(uintptr_t)&sB[1][0];

    // A tiles (16x4 f32, K padded with 1.0): lanes 0-15 hold K=0,1; 16-31 hold K=2,3
    const int qa0 = qbase + lr;
    const int qa1 = qbase + 16 + lr;
    float a0x, a0y, a1x, a1y;
    if (hh == 0) {
        a0x = xyz[3 * qa0];     a0y = xyz[3 * qa0 + 1];
        a1x = xyz[3 * qa1];     a1y = xyz[3 * qa1 + 1];
    } else {
        a0x = xyz[3 * qa0 + 2]; a0y = 1.0f;
        a1x = xyz[3 * qa1 + 2]; a1y = 1.0f;
    }
    v2f a0 = {a0x, a0y};
    v2f a1 = {a1x, a1y};

    // bounded top-54 (lane owns query qbase+lane)
    float kd[KNN];
    int   ki[KNN];
    for (int j = 0; j < KNN; ++j) { kd[j] = 3.4e38f; ki[j] = 0; }
    float worstd = 3.4e38f;
    int   worst  = 0;

    // prologue: DMA chunk 0 into buffer 0
    if (w == 0) tdm_load_chunk(bvec, lds_b0);

    for (int ci = 0; ci < NPTS / CHUNKC; ++ci) {
        const int  chunk    = ci * CHUNKC;
        const int  cb       = ci & 1;
        const bool has_next = (chunk + CHUNKC) < NPTS;
        if (w == 0) {
            if (has_next)
                tdm_load_chunk(bvec + chunk + CHUNKC, cb ? lds_b0 : lds_b1);
            // in-order TENSORcnt: <=1 outstanding means current chunk resident
            if (has_next) __builtin_amdgcn_s_wait_tensorcnt(1);
            else          __builtin_amdgcn_s_wait_tensorcnt(0);
        }
        __syncthreads();                 // chunk cb visible to all waves

        for (int tt = 0; tt < CHUNKC / 16; ++tt) {
            // B tile (4x16): lanes 0-15 read rows K=0,1; lanes 16-31 rows K=2,3
            const float2* bp = (const float2*)&sB[cb][tt * 16 + lr];
            float2 bl = bp[hh];
            v2f b = {bl.x, bl.y};

            v8f c0 = {};
            v8f c1 = {};
            c0 = __builtin_amdgcn_wmma_f32_16x16x4_f32(false, a0, false, b,
                                                       (short)0, c0, false, false);
            c1 = __builtin_amdgcn_wmma_f32_16x16x4_f32(false, a1, false, b,
                                                       (short)0, c1, false, false);

            // C layout: lane holds C[M = v + 8*hh][N = lr] in VGPR v.
            // Transpose into sD[w][n=N][m=M (+16 for tile 1)].
            const int mb = hh * 8;
            float* dst = &sD[w][lr][mb];
            *(float4*)(dst)      = make_float4(c0[0], c0[1], c0[2], c0[3]);
            *(float4*)(dst + 4)  = make_float4(c0[4], c0[5], c0[6], c0[7]);
            *(float4*)(dst + 16) = make_float4(c1[0], c1[1], c1[2], c1[3]);
            *(float4*)(dst + 20) = make_float4(c1[4], c1[5], c1[6], c1[7]);
            asm volatile("s_wait_dscnt 0" ::: "memory");   // intra-wave LDS RAW

            const int cbase = chunk + tt * 16;
            for (int n = 0; n < 16; ++n) {
                float d = sD[w][n][lane];
                if (d < worstd) {
                    kd[worst] = d;
                    ki[worst] = cbase + n;
                    worstd = kd[0]; worst = 0;
                    for (int j = 1; j < KNN; ++j)
                        if (kd[j] > worstd) { worstd = kd[j]; worst = j; }
                }
            }
        }
        __syncthreads();                 // all waves done with buffer cb before
    }                                    // next iteration's TDM overwrites it

    // sort ascending by distance (matches reference's resort-by-r)
    for (int i = 1; i < KNN; ++i) {
        float dv = kd[i]; int iv = ki[i]; int j = i;
        while (j > 0 && kd[j - 1] > dv) { kd[j] = kd[j - 1]; ki[j] = ki[j - 1]; --j; }
        kd[j] = dv; ki[j] = iv;
    }
    const int q = qbase + lane;
    for (int j = 0; j < KNN; ++j) knn_out[q * KNN + j] = ki[j];
}

// ---------------------------------------------------------------------------
// Kernel 3: umbrella features + h = feat@W1^T + b1, plus per-block partial
// sums of h and h^2 for the BatchNorm statistics (deterministic reduction).
// One thread per query point.
// ---------------------------------------------------------------------------
__global__ __launch_bounds__(256) void feat_kernel(const float* __restrict__ xyz,
                                                   const int* __restrict__ knn,
                                                   const float* __restrict__ W1,
                                                   const float* __restrict__ b1,
                                                   float* __restrict__ hbuf,
                                                   float* __restrict__ partial) {
    __shared__ float sW1[100];
    __shared__ float sB1[10];
    __shared__ float sred[256];
    const int tid = threadIdx.x;
    if (tid < 100) sW1[tid] = W1[tid];
    if (tid < 10)  sB1[tid] = b1[tid];
    __syncthreads();

    const int q = blockIdx.x * 256 + tid;
    const float qx = xyz[3 * q], qy = xyz[3 * q + 1], qz = xyz[3 * q + 2];

    float gx[KNN], gy[KNN], gz[KNN];
    for (int j = 0; j < KNN; ++j) {
        int i = knn[q * KNN + j];
        gx[j] = xyz[3 * i]     - qx;
        gy[j] = xyz[3 * i + 1] - qy;
        gz[j] = xyz[3 * i + 2] - qz;
    }

    // parts [0,9) [9,27) [27,54): sort each by phi of ROT-rotated offsets
    const int pb0[4] = {0, 9, 27, 54};
    for (int p = 0; p < 3; ++p) {
        const int s = pb0[p], e = pb0[p + 1];
        float ph[27];
        for (int j = s; j < e; ++j) {
            // rotated = g @ ROT ; ROT rows: [.5,-.5,.7071],[.7071,.7071,0],[-.5,.5,.7071]
            float rx = gx[j] * 0.5f     + gy[j] * 0.7071f + gz[j] * (-0.5f);
            float ry = gx[j] * (-0.5f)  + gy[j] * 0.7071f + gz[j] * 0.5f;
            ph[j - s] = atan2f(ry, rx);          // monotone in phi
        }
        for (int i2 = s + 1; i2 < e; ++i2) {      // stable insertion sort
            float pv = ph[i2 - s], x = gx[i2], y = gy[i2], z = gz[i2];
            int j2 = i2;
            while (j2 > s && ph[j2 - 1 - s] > pv) {
                ph[j2 - s] = ph[j2 - 1 - s];
                gx[j2] = gx[j2 - 1]; gy[j2] = gy[j2 - 1]; gz[j2] = gz[j2 - 1];
                --j2;
            }
            ph[j2 - s] = pv; gx[j2] = x; gy[j2] = y; gz[j2] = z;
        }
    }

    // umbrella triangles: v0=0, v1=s[i], v2=s[(i+1)%L]
    float ux[KNN], uy[KNN], uz[KNN], cx[KNN], cy[KNN], cz[KNN];
    unsigned long long badm = 0ull;
    for (int p = 0; p < 3; ++p) {
        const int s = pb0[p], L = pb0[p + 1] - pb0[p];
        for (int i2 = 0; i2 < L; ++i2) {
            const int t  = s + i2;
            const int t2 = s + ((i2 + 1) % L);
            float e1x = gx[t],  e1y = gy[t],  e1z = gz[t];
            float e2x = gx[t2], e2y = gy[t2], e2z = gz[t2];
            float nx = e1y * e2z - e1z * e2y;
            float ny = e1z * e2x - e1x * e2z;
            float nz = e1x * e2y - e1y * e2x;
            float nl = sqrtf(nx * nx + ny * ny + nz * nz);
            if (nl == 0.f) { badm |= (1ull << t); ux[t] = uy[t] = uz[t] = 0.f; }
            else { float r = 1.f / nl; ux[t] = nx * r; uy[t] = ny * r; uz[t] = nz * r; }
            cx[t] = (e1x + e2x) * (1.f / 3.f);
            cy[t] = (e1y + e2y) * (1.f / 3.f);
            cz[t] = (e1z + e2z) * (1.f / 3.f);
        }
    }
    const float sign = (ux[0] > 0.f) ? 1.f : -1.f;   // global per-point sign
    int first = 0;
    for (int t = 0; t < KNN; ++t)
        if (!((badm >> t) & 1ull)) { first = t; break; }

    float sh[10], sh2[10];
    for (int c = 0; c < 10; ++c) { sh[c] = 0.f; sh2[c] = 0.f; }

    for (int t = 0; t < KNN; ++t) {
        // polar from ORIGINAL cen (pre-replacement, as in reference)
        float px = cx[t], py = cy[t], pz = cz[t];
        float rho = sqrtf(px * px + py * py + pz * pz);
        float theta = 0.f;
        if (rho != 0.f) {
            float cc = pz / rho;
            cc = fminf(1.f, fmaxf(-1.f, cc));
            theta = acosf(cc) * (1.f / PI_F);
        }
        float phi = atan2f(py, px) * INV2PI_F + 0.5f;

        const int tt = ((badm >> t) & 1ull) ? first : t;
        float nx = ux[tt] * sign, ny = uy[tt] * sign, nz = uz[tt] * sign;
        float ccx = cx[tt], ccy = cy[tt], ccz = cz[tt];
        float pos = (nx * ccx + ny * ccy + nz * ccz) * (1.f / 1.7320508075688772f);

        float f[10] = {rho, theta, phi, nx, ny, nz, pos, ccx, ccy, ccz};
        float* hp = &hbuf[(size_t)q * 540 + t * 10];
        for (int c = 0; c < 10; ++c) {
            float h = sB1[c];
            for (int k2 = 0; k2 < 10; ++k2) h += sW1[c * 10 + k2] * f[k2];
            hp[c] = h;
            sh[c]  += h;
            sh2[c] += h * h;
        }
    }

    // deterministic block tree-reduction of 20 channels (sum_h, sum_h2)
    for (int c = 0; c < 20; ++c) {
        float v = (c < 10) ? sh[c] : sh2[c - 10];
        __syncthreads();
        sred[tid] = v;
        __syncthreads();
        for (int off = 128; off > 0; off >>= 1) {
            if (tid < off) sred[tid] += sred[tid + off];
            __syncthreads();
        }
        if (tid == 0) partial[blockIdx.x * 20 + c] = sred[0];
    }
}

// ---------------------------------------------------------------------------
// Kernel 4: finalize BN stats -> per-channel scale/shift (fixed-order reduce)
// ---------------------------------------------------------------------------
__global__ void stats_kernel(const float* __restrict__ partial,
                             const float* __restrict__ gamma,
                             const float* __restrict__ beta,
                             float* __restrict__ bn) {
    int c = threadIdx.x;
    if (c >= 10) return;
    float s = 0.f, s2 = 0.f;
    for (int b = 0; b < 256; ++b) {
        s  += partial[b * 20 + c];
        s2 += partial[b * 20 + 10 + c];
    }
    const float inv = 1.f / (65536.f * 54.f);
    float mu  = s * inv;
    float var = s2 * inv - mu * mu;
    float rs  = rsqrtf(var + 1e-5f);
    float sc  = gamma[c] * rs;
    bn[c]      = sc;
    bn[10 + c] = beta[c] - mu * sc;
}

// ---------------------------------------------------------------------------
// Kernel 5: BN + ReLU + second linear + sum over the 54 triangles.
// out[q][co] = sum_ci W2[co][ci] * (sum_t relu(bn(h))) + 54*b2[co]
// ---------------------------------------------------------------------------
__global__ __launch_bounds__(256) void out_kernel(const float* __restrict__ hbuf,
                                                  const float* __restrict__ bn,
                                                  const float* __restrict__ W2,
                                                  const float* __restrict__ b2,
                                                  float* __restrict__ out) {
    const int q = blockIdx.x * 256 + threadIdx.x;
    float sc[10], sf[10], acc[10];
    for (int c = 0; c < 10; ++c) { sc[c] = bn[c]; sf[c] = bn[10 + c]; acc[c] = 0.f; }
    const float* hp = &hbuf[(size_t)q * 540];
    for (int t = 0; t < KNN; ++t)
        for (int c = 0; c < 10; ++c) {
            float h = hp[t * 10 + c] * sc[c] + sf[c];
            acc[c] += fmaxf(h, 0.f);
        }
    for (int co = 0; co < 10; ++co) {
        float o = 54.f * b2[co];
        for (int ci = 0; ci < 10; ++ci) o += W2[co * 10 + ci] * acc[ci];
        out[q * 10 + co] = o;
    }
}

// ---------------------------------------------------------------------------
extern "C" void kernel_launch(void* const* d_in, const int* in_sizes, int n_in,
                              void* d_out, int out_size, void* d_ws, size_t ws_size,
                              hipStream_t stream) {
    const float* center = (const float*)d_in[0];
    // d_in[1] = offset (unused)
    const float* W1    = (const float*)d_in[2];
    const float* b1    = (const float*)d_in[3];
    const float* gamma = (const float*)d_in[4];
    const float* beta  = (const float*)d_in[5];
    const float* W2    = (const float*)d_in[6];
    const float* b2    = (const float*)d_in[7];

    char* w = (char*)d_ws;
    float4* bvec   = (float4*)w;                                     //   1.0 MB
    int*    idx    = (int*)(w + (size_t)(1 << 20));                  //  14.2 MB
    float*  hbuf   = (float*)(w + (size_t)(16 << 20));               // 141.6 MB
    float*  partial= (float*)(w + (size_t)(16 << 20) + 141557760ull);//  20 KB
    float*  bn     = partial + 256 * 20;                             //  80 B
    float*  out    = (float*)d_out;

    bvec_kernel <<<NPTS / 256, 256, 0, stream>>>(center, bvec);
    knn_kernel  <<<NPTS / 256, 256, 0, stream>>>(center, bvec, idx);
    feat_kernel <<<NPTS / 256, 256, 0, stream>>>(center, idx, W1, b1, hbuf, partial);
    stats_kernel<<<1, 16, 0, stream>>>(partial, gamma, beta, bn);
    out_kernel  <<<NPTS / 256, 256, 0, stream>>>(hbuf, bn, W2, b2, out);
}